// RoutingLayer_63728724738084
// MI455X (gfx1250) — compile-verified
//
#include <hip/hip_runtime.h>
#include <hip/hip_bf16.h>
#include <stdint.h>

#define DIM     128
#define ROUTIT  6
#define TILE    16      // edges per TDM gather descriptor (16 x 16-bit row indices max)
#define WPB     4       // waves per block

typedef uint32_t v4u __attribute__((ext_vector_type(4)));
typedef int      v8i __attribute__((ext_vector_type(8)));
typedef int      v4i __attribute__((ext_vector_type(4)));

#if __has_builtin(__builtin_amdgcn_tensor_load_to_lds) && __has_builtin(__builtin_amdgcn_s_wait_tensorcnt)
#define HAVE_TDM 1
#else
#define HAVE_TDM 0
#endif

// ---- wave32 cross-lane helpers: single ds_swizzle / ds_bpermute, no index math ----
#define XOR1  0x041F
#define XOR2  0x081F
#define XOR4  0x101F
#define XOR8  0x201F
#define XOR16 0x401F

template <int PAT>
__device__ __forceinline__ float xor_addf(float v)
{
    return v + __int_as_float(__builtin_amdgcn_ds_swizzle(__float_as_int(v), PAT));
}
__device__ __forceinline__ float bpermf(int byteIdx, float v)
{
    return __int_as_float(__builtin_amdgcn_ds_bpermute(byteIdx, __float_as_int(v)));
}

// ---------------------------------------------------------------------------
// Capsule L2-normalize x -> xn. 4 lanes per capsule (one float4 each).
// ---------------------------------------------------------------------------
__global__ void caps_norm_x(const float* __restrict__ x, float* __restrict__ xn, int ncap4)
{
    int t = blockIdx.x * blockDim.x + threadIdx.x;
    if (t >= ncap4) return;
    float4 v = ((const float4*)x)[t];
    float ss = v.x * v.x + v.y * v.y + v.z * v.z + v.w * v.w;
    ss = xor_addf<XOR1>(ss);
    ss = xor_addf<XOR2>(ss);
    float inv = 1.0f / fmaxf(sqrtf(ss), 1e-12f);
    v.x *= inv; v.y *= inv; v.z *= inv; v.w *= inv;
    ((float4*)xn)[t] = v;
}

__global__ void zero_f4(float* __restrict__ p, int n4)
{
    int t = blockIdx.x * blockDim.x + threadIdx.x;
    if (t < n4) ((float4*)p)[t] = make_float4(0.f, 0.f, 0.f, 0.f);
}

// out = S + xn, capsule-normalized unless last iteration
__global__ void caps_combine(const float* __restrict__ S, const float* __restrict__ xn,
                             float* __restrict__ out, int ncap4, int do_norm)
{
    int t = blockIdx.x * blockDim.x + threadIdx.x;
    if (t >= ncap4) return;
    float4 a = ((const float4*)S)[t];
    float4 b = ((const float4*)xn)[t];
    float4 v = make_float4(a.x + b.x, a.y + b.y, a.z + b.z, a.w + b.w);
    if (do_norm) {
        float ss = v.x * v.x + v.y * v.y + v.z * v.z + v.w * v.w;
        ss = xor_addf<XOR1>(ss);
        ss = xor_addf<XOR2>(ss);
        float inv = 1.0f / fmaxf(sqrtf(ss), 1e-12f);
        v.x *= inv; v.y *= inv; v.z *= inv; v.w *= inv;
    }
    ((float4*)out)[t] = v;
}

// ---------------------------------------------------------------------------
// Per-edge routing math. Lane L owns elements 4L..4L+3 of the 128-wide row.
//   s[j]   = sum of 8 consecutive c elems (lanes 2j,2j+1 after one xor1 add)
//   p[i]   = sum_j z[i,j]*s[j]           (xor1+xor2 reduce within 4-lane group)
//   softmax over 8 capsules: |p|<=32 since rows are unit-norm -> exp safe
//   without max-shift (softmax is shift-invariant); xor4/8/16 sum reduce.
// ---------------------------------------------------------------------------
__device__ __forceinline__ void route_edge(float4 z4, float4 c4, float* dst, int lane)
{
    float part = c4.x + c4.y + c4.z + c4.w;
    float sj = xor_addf<XOR1>(part);              // lanes 2j,2j+1 hold s[j]
    int bidx = (lane & 3) << 5;                   // byte idx of lane 2*j0, j0=4*(lane&3)
    float s0 = bpermf(bidx + 0,  sj);
    float s1 = bpermf(bidx + 8,  sj);
    float s2 = bpermf(bidx + 16, sj);
    float s3 = bpermf(bidx + 24, sj);
    float pp = fmaf(z4.x, s0, fmaf(z4.y, s1, fmaf(z4.z, s2, z4.w * s3)));
    pp = xor_addf<XOR1>(pp);
    pp = xor_addf<XOR2>(pp);                      // lane L holds p[L>>2]
    float ex = __expf(pp);
    float sm = xor_addf<XOR4>(ex);
    sm = xor_addf<XOR8>(sm);
    sm = xor_addf<XOR16>(sm);
    float w = ex / sm;
    unsafeAtomicAdd(dst + 0, w * z4.x);
    unsafeAtomicAdd(dst + 1, w * z4.y);
    unsafeAtomicAdd(dst + 2, w * z4.z);
    unsafeAtomicAdd(dst + 3, w * z4.w);
}

#if HAVE_TDM
// Issue one TDM gather-mode descriptor: 16 indexed rows of 128 f32 -> LDS tile.
__device__ __forceinline__ void tdm_issue_gather16(const float* base, uint32_t lds_addr,
                                                   const int* __restrict__ idx,
                                                   long e0, int count, int n)
{
    uint32_t id[TILE];
#pragma unroll
    for (int k = 0; k < TILE; ++k) {
        long e = (k < count) ? (e0 + k) : (e0 + count - 1);   // repeats are legal
        id[k] = (uint32_t)idx[e];
    }
    uint64_t ga = (uint64_t)(uintptr_t)base;
    v4u g0;
    g0.x = 0x80000001u;                                       // gather_mode, 16b idx, count=1
    g0.y = lds_addr;
    g0.z = (uint32_t)ga;
    g0.w = (uint32_t)((ga >> 32) & 0x01FFFFFFu) | 0x80000000u; // addr[56:32] | type=2
    v8i g1;
    g1[0] = 0x00020000;                                        // data_size = 4B
    g1[1] = (int)((uint32_t)DIM << 16);                        // tensor_dim0 = 128
    g1[2] = (int)(((uint32_t)n & 0xFFFFu) << 16);              // tensor_dim1 lo
    g1[3] = (int)((((uint32_t)n >> 16) & 0xFFFFu) | ((uint32_t)DIM << 16)); // dim1 hi | tile_dim0
    g1[4] = count;                                             // tile_dim1 = #valid indices
    g1[5] = DIM;                                               // tensor_dim0_stride = 128
    g1[6] = 0;
    g1[7] = 0;
    v4i g2, g3;
#pragma unroll
    for (int k = 0; k < 4; ++k) {
        g2[k] = (int)((id[2 * k] & 0xFFFFu) | ((id[2 * k + 1] & 0xFFFFu) << 16));
        g3[k] = (int)((id[8 + 2 * k] & 0xFFFFu) | ((id[8 + 2 * k + 1] & 0xFFFFu) << 16));
    }
#if __clang_major__ >= 23
    v8i gz = {0, 0, 0, 0, 0, 0, 0, 0};
    __builtin_amdgcn_tensor_load_to_lds(g0, g1, g2, g3, gz, 0);
#else
    __builtin_amdgcn_tensor_load_to_lds(g0, g1, g2, g3, 0);
#endif
}
#endif // HAVE_TDM

// ---------------------------------------------------------------------------
// Edge kernel: grid-stride over 16-edge tiles, double-buffered TDM gathers.
// Issue tile t+1's two gathers, s_wait_tensorcnt(2) -> tile t's pair is done
// (per-wave tensor ops complete in order), compute tile t while DMA fills the
// other buffer.
// ---------------------------------------------------------------------------
__global__ __launch_bounds__(32 * WPB)
void edge_route(const float* __restrict__ xn, const float* __restrict__ cin,
                const int* __restrict__ src, const int* __restrict__ trg,
                float* __restrict__ S, int m, int n, int ntiles)
{
    const int lane = threadIdx.x & 31;
    const int wv   = threadIdx.x >> 5;
    const int wave = blockIdx.x * WPB + wv;
    const int nwav = gridDim.x * WPB;

#if HAVE_TDM
    __shared__ float lz[WPB][2][TILE * DIM];
    __shared__ float lc[WPB][2][TILE * DIM];

    if (wave < ntiles) {
        long e0 = (long)wave * TILE;
        int cnt = ((long)m - e0 < TILE) ? (int)((long)m - e0) : TILE;
        tdm_issue_gather16(cin, (uint32_t)(uintptr_t)&lc[wv][0][0], trg, e0, cnt, n);
        tdm_issue_gather16(xn,  (uint32_t)(uintptr_t)&lz[wv][0][0], src, e0, cnt, n);
    }
    int pb = 0;
    for (int t = wave; t < ntiles; t += nwav, pb ^= 1) {
        long e0 = (long)t * TILE;
        int cnt = ((long)m - e0 < TILE) ? (int)((long)m - e0) : TILE;
        int tn = t + nwav;
        if (tn < ntiles) {
            long ne0 = (long)tn * TILE;
            int ncnt = ((long)m - ne0 < TILE) ? (int)((long)m - ne0) : TILE;
            tdm_issue_gather16(cin, (uint32_t)(uintptr_t)&lc[wv][pb ^ 1][0], trg, ne0, ncnt, n);
            tdm_issue_gather16(xn,  (uint32_t)(uintptr_t)&lz[wv][pb ^ 1][0], src, ne0, ncnt, n);
            __builtin_amdgcn_s_wait_tensorcnt(2);   // current tile's pair complete
        } else {
            __builtin_amdgcn_s_wait_tensorcnt(0);
        }
#pragma unroll 4
        for (int sub = 0; sub < TILE; ++sub) {
            if (sub >= cnt) break;
            int tv = trg[e0 + sub];
            float4 z4 = *(const float4*)&lz[wv][pb][sub * DIM + lane * 4];
            float4 c4 = *(const float4*)&lc[wv][pb][sub * DIM + lane * 4];
            route_edge(z4, c4, S + (size_t)tv * DIM + lane * 4, lane);
        }
    }
#else
    for (int t = wave; t < ntiles; t += nwav) {
        long e0 = (long)t * TILE;
        int cnt = ((long)m - e0 < TILE) ? (int)((long)m - e0) : TILE;
#pragma unroll 4
        for (int sub = 0; sub < TILE; ++sub) {
            if (sub >= cnt) break;
            int sv = src[e0 + sub];
            int tv = trg[e0 + sub];
            float4 z4 = ((const float4*)xn)[(size_t)sv * (DIM / 4) + lane];
            float4 c4 = ((const float4*)cin)[(size_t)tv * (DIM / 4) + lane];
            route_edge(z4, c4, S + (size_t)tv * DIM + lane * 4, lane);
        }
    }
#endif
}

// ---------------------------------------------------------------------------
extern "C" void kernel_launch(void* const* d_in, const int* in_sizes, int n_in,
                              void* d_out, int out_size, void* d_ws, size_t ws_size,
                              hipStream_t stream)
{
    const float* x  = (const float*)d_in[0];
    const int*   ei = (const int*)d_in[1];     // int32 edge_index (JAX x64 disabled)
    const int n = in_sizes[0] / DIM;
    const int m = in_sizes[1] / 2;
    const int* src = ei;
    const int* trg = ei + m;

    float* xn  = (float*)d_ws;                 // [n,128] normalized x
    float* S   = xn + (size_t)n * DIM;         // [n,128] scatter accumulator
    float* out = (float*)d_out;                // c lives here across iterations

    const int ncap4 = n * (DIM / 4);
    dim3 blk(256);
    dim3 grd((unsigned)((ncap4 + 255) / 256));

    caps_norm_x<<<grd, blk, 0, stream>>>(x, xn, ncap4);

    const int ntiles = (m + TILE - 1) / TILE;
    long eblocks = ((long)ntiles + WPB - 1) / WPB;
    if (eblocks > 2048) eblocks = 2048;        // grid-stride; ~2 tiles in flight per wave
    if (eblocks < 1) eblocks = 1;
    dim3 eblk(32 * WPB);
    dim3 egrd((unsigned)eblocks);

    for (int t = 0; t < ROUTIT; ++t) {
        zero_f4<<<grd, blk, 0, stream>>>(S, ncap4);
        const float* cin = (t == 0) ? xn : out;
        edge_route<<<egrd, eblk, 0, stream>>>(xn, cin, src, trg, S, m, n, ntiles);
        caps_combine<<<grd, blk, 0, stream>>>(S, xn, out, ncap4, (t < ROUTIT - 1) ? 1 : 0);
    }
}